// ResampleOctavePyTorch_66529043415273
// MI455X (gfx1250) — compile-verified
//
#include <hip/hip_runtime.h>
#include <hip/hip_bf16.h>

typedef float v2f __attribute__((ext_vector_type(2)));
typedef float v8f __attribute__((ext_vector_type(8)));

// Problem constants (reference harness: 48 kHz -> 10 kHz, 481-tap FIR)
#define UP      5
#define DOWN    24
#define HLEN    481      // filter taps
#define MSHIFT  10       // n_pre_remove = ((HLEN-1)/2)/DOWN  (n_pre_pad == 0)
#define JMAX    96       // (HLEN-1)/UP
#define KSTEPS  43       // ceil(170/4): dense K window per 16-output tile
#define XCOLS   712      // staged input window columns per workgroup
#define XSTRIDE 721      // LDS row stride (odd -> conflict-free strided reads)
#define YSTRIDE 132      // LDS output stride (132 % 64 == 4 -> conflict-free)
#define TAPMAX  844      // max tap index reached (24*15 + 4 + 480)
#define HPAD    1232     // reversed padded table size (>= 844+375+1)

__global__ __launch_bounds__(256) void resample_poly_wmma(
    const float* __restrict__ x,   // [B][n_in]
    const float* __restrict__ h,   // [HLEN]
    float* __restrict__ out,       // [B][n_out]
    int n_in, int n_out)
{
    __shared__ float hrev[HPAD];              // REVERSED zero-padded 5*h
    __shared__ float xs[16 * XSTRIDE];        // 16 batches x input window
    __shared__ float yo[16 * YSTRIDE];        // 16 batches x 128 outputs

    const int tid  = threadIdx.x;
    const int wave = tid >> 5;
    const int lane = tid & 31;
    const int b0    = blockIdx.y << 4;        // batch tile base
    const int m_wg0 = blockIdx.x << 7;        // 128 outputs per workgroup

    // ---- stage reversed, zero-padded, scaled filter ----
    // hrev[j] = 5*h[TAPMAX - j] inside the band, else 0  (taps span [-375, 844])
    for (int i = tid; i < HPAD; i += 256) {
        int k = TAPMAX - i;
        hrev[i] = ((unsigned)k < (unsigned)HLEN) ? 5.0f * h[k] : 0.0f;
    }

    // ---- stage input window: 16 batch rows, coalesced per row ----
    const int t0_wg = (DOWN * (m_wg0 + MSHIFT)) / UP;
    const int t_min = t0_wg - JMAX;                     // negative only at m=0
    for (int r = 0; r < 16; ++r) {
        const float* xrow = x + (size_t)(b0 + r) * (size_t)n_in;
        float* dst = &xs[r * XSTRIDE];
        for (int c = tid; c < XCOLS; c += 256) {
            int t = t_min + c;
            dst[c] = ((unsigned)t < (unsigned)n_in) ? xrow[t] : 0.0f;
        }
    }
    __syncthreads();

    // ---- per-wave 16x16 tile (M = outputs, N = batches, K = input samples) ----
    int tile = (blockIdx.x << 3) + wave;
    const int last_tile = (n_out >> 4) - 1;
    if (tile > last_tile) tile = last_tile;             // uniform clamp, EXEC stays full
    const int m0   = tile << 4;
    const int t0t  = (DOWN * (m0 + MSHIFT)) / UP;
    const int toff = (t0t - JMAX) - t_min;              // column offset into xs

    const int mrow = lane & 15;                         // A row (M); B col (N = batch)
    const int ksub = (lane >> 4) << 1;                  // 0 | 2 (K sub-pair per half-wave)
    // tap(m,K) = DOWN*(m0+m+MSHIFT) - UP*(tbase+K); tap at K=ksub:
    const int tap0 = DOWN * (m0 + mrow + MSHIFT) - UP * (t0t - JMAX) - UP * ksub;

    // reversed-table base: av0 index ascends with kk and is BELOW av1 -> no reg swaps
    const float* hA = &hrev[TAPMAX - tap0];             // index >= 0
    const float* xB = &xs[mrow * XSTRIDE + toff + ksub];

    v8f acc = {0.f, 0.f, 0.f, 0.f, 0.f, 0.f, 0.f, 0.f};

    #pragma unroll
    for (int kk = 0; kk < KSTEPS; ++kk) {
        // A fragment: {K, K+1} taps -> ascending reversed-table offsets {20kk, 20kk+5}
        float av0 = hA[20 * kk];                        // tap = tap0 - 20kk
        float av1 = hA[20 * kk + 5];                    // tap = tap0 - 20kk - 5
        // B fragment: input samples, conflict-free strided LDS gather
        float bv0 = xB[4 * kk];
        float bv1 = xB[4 * kk + 1];
        v2f A = {av0, av1};
        v2f B = {bv0, bv1};
        acc = __builtin_amdgcn_wmma_f32_16x16x4_f32(
            false, A, false, B, (short)0, acc, false, false);
    }

    // ---- C layout: vgpr i -> M = i + 8*(lane>>4), N = lane&15 ----
    {
        const int colw = (wave << 4) + ((lane >> 4) << 3);
        float* yrow = &yo[mrow * YSTRIDE + colw];
        #pragma unroll
        for (int i = 0; i < 8; ++i) yrow[i] = acc[i];
    }
    __syncthreads();

    // ---- coalesced global store: 128 contiguous outputs per batch row ----
    for (int idx = tid; idx < 16 * 128; idx += 256) {
        int r = idx >> 7;
        int c = idx & 127;
        int m = m_wg0 + c;
        if (m < n_out)
            out[(size_t)(b0 + r) * (size_t)n_out + m] = yo[r * YSTRIDE + c];
    }
}

extern "C" void kernel_launch(void* const* d_in, const int* in_sizes, int n_in_bufs,
                              void* d_out, int out_size, void* d_ws, size_t ws_size,
                              hipStream_t stream) {
    const float* x = (const float*)d_in[0];
    const float* h = (const float*)d_in[1];
    float* out = (float*)d_out;

    const int T = 1440000;                 // samples per batch (reference harness)
    const int B = in_sizes[0] / T;         // = 32
    const int n_out = (int)(((long long)T * UP) / DOWN); // 300000, exact

    dim3 grid((n_out + 127) / 128, B / 16);
    resample_poly_wmma<<<grid, 256, 0, stream>>>(x, h, out, T, n_out);
}